// lgc_loss_79628693667852
// MI455X (gfx1250) — compile-verified
//
#include <hip/hip_runtime.h>
#include <hip/hip_bf16.h>
#include <math.h>

typedef __attribute__((ext_vector_type(2))) float v2f;
typedef __attribute__((ext_vector_type(8))) float v8f;

#define NROWS 65536
#define KCOLS 256
#define DDIM  256
#define TK    5
#define FBIG  3.4e38f

static __device__ __forceinline__ float waveReduceSum(float v) {
  #pragma unroll
  for (int off = 16; off > 0; off >>= 1) v += __shfl_xor(v, off, 32);
  return v;
}

static __device__ __forceinline__ float blockReduceSum256(float v) {
  __shared__ float sm[8];
  v = waveReduceSum(v);
  const int lane = threadIdx.x & 31, wave = threadIdx.x >> 5;
  if (lane == 0) sm[wave] = v;
  __syncthreads();
  if (wave == 0) {
    float r = (lane < 8) ? sm[lane] : 0.0f;
    r = waveReduceSum(r);
    if (lane == 0) sm[0] = r;
  }
  __syncthreads();
  return sm[0];
}

static __device__ __forceinline__ void insert5(float* a, float v) {
  if (v >= a[4]) return;
  a[4] = v;
  #pragma unroll
  for (int t = 4; t > 0; --t)
    if (a[t] < a[t-1]) { float tmp = a[t]; a[t] = a[t-1]; a[t-1] = tmp; }
}

static __device__ __forceinline__ void merge5(float* a, const float* b) {
  float o[5]; int ia = 0, ib = 0;
  #pragma unroll
  for (int t = 0; t < 5; ++t) {
    float av = a[ia], bv = b[ib];
    if (av <= bv) { o[t] = av; ++ia; } else { o[t] = bv; ++ib; }
  }
  #pragma unroll
  for (int t = 0; t < 5; ++t) a[t] = o[t];
}

// ---- row stats: sum(x^2), sum(x) per row (one wave per row) ----
__global__ void __launch_bounds__(256) stats_kernel(const float* __restrict__ x,
                                                    float2* __restrict__ st, int nrows) {
  const int row  = blockIdx.x * 8 + (threadIdx.x >> 5);
  const int lane = threadIdx.x & 31;
  if (row >= nrows) return;
  const float* p = x + (size_t)row * DDIM;
  float s = 0.0f, q = 0.0f;
  #pragma unroll
  for (int t = 0; t < 8; ++t) { float v = p[lane + t * 32]; s += v; q += v * v; }
  q = waveReduceSum(q);
  s = waveReduceSum(s);
  if (lane == 0) st[row] = make_float2(q, s);
}

// ---- pairwise distance via fp32 WMMA 16x16x4 ----
__global__ void __launch_bounds__(256) dist_wmma_kernel(const float* __restrict__ z,
                                                        const float* __restrict__ mu,
                                                        const float2* __restrict__ zs,
                                                        const float2* __restrict__ ms,
                                                        float* __restrict__ dbuf) {
  __shared__ __align__(16) float zt[16 * DDIM];
  const int i0 = blockIdx.x * 16;
  { // stage 16x256 z tile into LDS, coalesced b128 loads
    const float4* src = (const float4*)(z + (size_t)i0 * DDIM);
    float4* dst = (float4*)zt;
    #pragma unroll
    for (int r = 0; r < 4; ++r) dst[r * 256 + threadIdx.x] = src[r * 256 + threadIdx.x];
  }
  __syncthreads();

  const int lane = threadIdx.x & 31;
  const int wave = threadIdx.x >> 5;
  const int m    = lane & 15;
  const int kh   = (lane >> 4) * 2;      // 0 or 2 (K slot within the 4-chunk)

  for (int jt = 0; jt < 2; ++jt) {
    const int j0 = (wave * 2 + jt) * 16;
    v8f acc = {};
    const float* zrow = zt + m * DDIM + kh;
    const float* mrow = mu + (size_t)(j0 + m) * DDIM + kh;
    #pragma unroll 4
    for (int kk = 0; kk < DDIM; kk += 4) {
      v2f a, b;
      a.x = zrow[kk]; a.y = zrow[kk + 1];
      b.x = mrow[kk]; b.y = mrow[kk + 1];
      acc = __builtin_amdgcn_wmma_f32_16x16x4_f32(false, a, false, b,
                                                  (short)0, acc, false, false);
    }
    // epilogue: d = sqrt(max(z2 + m2 - 2*cross + 2*eps*(zsum-musum) + D*eps^2, 0))
    const int j = j0 + m;
    const float2 msj = ms[j];
    const int mbase = (lane >> 4) * 8;
    #pragma unroll
    for (int v = 0; v < 8; ++v) {
      const int i = i0 + mbase + v;
      const float2 zsi = zs[i];
      float d2 = zsi.x + msj.x - 2.0f * acc[v]
               + 2.0e-6f * (zsi.y - msj.y) + 2.56e-10f;
      dbuf[(size_t)i * KCOLS + j] = sqrtf(fmaxf(d2, 0.0f));
    }
  }
}

// ---- per-row top-5 smallest (one wave per row) ----
__global__ void __launch_bounds__(256) rowtopk_kernel(const float* __restrict__ dbuf,
                                                      float* __restrict__ relz) {
  const int row  = blockIdx.x * 8 + (threadIdx.x >> 5);
  const int lane = threadIdx.x & 31;
  const float* p = dbuf + (size_t)row * KCOLS;
  float a[5] = {FBIG, FBIG, FBIG, FBIG, FBIG};
  #pragma unroll
  for (int t = 0; t < 8; ++t) insert5(a, p[lane + t * 32]);
  #pragma unroll
  for (int off = 16; off > 0; off >>= 1) {
    float b[5];
    #pragma unroll
    for (int t = 0; t < 5; ++t) b[t] = __shfl_xor(a[t], off, 32);
    merge5(a, b);
  }
  if (lane == 0) {
    #pragma unroll
    for (int t = 0; t < 5; ++t) relz[(size_t)row * TK + t] = a[t];
  }
}

// ---- per-column top-5: coalesced partials over row chunks ----
__global__ void __launch_bounds__(256) colpart_kernel(const float* __restrict__ dbuf,
                                                      float* __restrict__ part) {
  const int j = threadIdx.x, b = blockIdx.x;
  float a[5] = {FBIG, FBIG, FBIG, FBIG, FBIG};
  const float* p = dbuf + (size_t)b * 256 * KCOLS + j;
  for (int r = 0; r < 256; ++r) insert5(a, p[(size_t)r * KCOLS]);
  #pragma unroll
  for (int t = 0; t < 5; ++t) part[((size_t)b * KCOLS + j) * TK + t] = a[t];
}

__global__ void __launch_bounds__(256) colmerge_kernel(const float* __restrict__ part,
                                                       float* __restrict__ relu_) {
  const int j = threadIdx.x;
  float a[5] = {FBIG, FBIG, FBIG, FBIG, FBIG};
  for (int b = 0; b < 256; ++b)
    #pragma unroll
    for (int t = 0; t < 5; ++t) insert5(a, part[((size_t)b * KCOLS + j) * TK + t]);
  #pragma unroll
  for (int t = 0; t < 5; ++t) relu_[(size_t)j * TK + t] = a[t];
}

// ---- sigma bisection (matches reference: sigma0=1, mid=[0,1e4], 150 iters) ----
__global__ void __launch_bounds__(256) sigma_kernel(const float* __restrict__ rel,
                                                    float* __restrict__ sigma,
                                                    float target, int n) {
  const int i = blockIdx.x * blockDim.x + threadIdx.x;
  if (i >= n) return;
  float r[5];
  #pragma unroll
  for (int t = 0; t < 5; ++t) r[t] = rel[(size_t)i * TK + t];
  const float rho = r[0];
  float lo = 0.0f, hi = 10000.0f, sg = 1.0f;
  for (int it = 0; it < TK * 30; ++it) {
    float cur = 0.0f;
    #pragma unroll
    for (int t = 0; t < 5; ++t) cur += __expf(-fmaxf(r[t] - rho, 0.0f) / sg);
    if (cur > target) hi = sg; else lo = sg;
    sg = 0.5f * (lo + hi);
  }
  sigma[i] = sg;
}

// ---- W1/W2/S + row-normalize (block = one row) ----
__global__ void __launch_bounds__(256) ws_kernel(const float* __restrict__ dbuf,
                                                 const float* __restrict__ relz,
                                                 const float* __restrict__ sigz,
                                                 const float* __restrict__ relu_,
                                                 const float* __restrict__ sigu,
                                                 float* __restrict__ out) {
  const size_t NK = (size_t)NROWS * KCOLS;
  const int i = blockIdx.x, j = threadIdx.x;
  const float rho_z = relz[(size_t)i * TK];
  const float sgz   = sigz[i];
  const float rho_u = relu_[(size_t)j * TK];
  const float sgu   = sigu[j];
  const float d = dbuf[(size_t)i * KCOLS + j];
  const float w1 = __expf(-fmaxf(d - rho_z, 0.0f) / sgz);
  const float w2 = __expf(-fmaxf(d - rho_u, 0.0f) / sgu);
  const float s  = w1 + w2 - w1 * w2;
  const float rs = blockReduceSum256(s);
  const float sn = s / rs;
  const size_t idx = (size_t)i * KCOLS + j;
  out[idx]          = w1;  // W1
  out[NK + idx]     = sn;  // S (row-normalized)
  out[2 * NK + idx] = w1;  // W1 again
}

// ---- column sums of normalized S ----
__global__ void __launch_bounds__(256) cspart_kernel(const float* __restrict__ sn,
                                                     float* __restrict__ part) {
  const int j = threadIdx.x, b = blockIdx.x;
  float acc = 0.0f;
  const float* p = sn + (size_t)b * 256 * KCOLS + j;
  for (int r = 0; r < 256; ++r) acc += p[(size_t)r * KCOLS];
  part[(size_t)b * KCOLS + j] = acc;
}

__global__ void __launch_bounds__(256) csmerge_kernel(const float* __restrict__ part,
                                                      float* __restrict__ colsum) {
  const int j = threadIdx.x;
  float acc = 0.0f;
  for (int b = 0; b < 256; ++b) acc += part[(size_t)b * KCOLS + j];
  colsum[j] = acc;
}

// ---- Dmat = rownorm( S^2 / colsum ) (block = one row) ----
__global__ void __launch_bounds__(256) dmat_kernel(const float* __restrict__ colsum,
                                                   float* __restrict__ out) {
  const size_t NK = (size_t)NROWS * KCOLS;
  const int i = blockIdx.x, j = threadIdx.x;
  const size_t idx = (size_t)i * KCOLS + j;
  const float s = out[NK + idx];
  const float v = s * s / colsum[j];
  const float rs = blockReduceSum256(v);
  out[3 * NK + idx] = v / rs;
}

extern "C" void kernel_launch(void* const* d_in, const int* in_sizes, int n_in,
                              void* d_out, int out_size, void* d_ws, size_t ws_size,
                              hipStream_t stream) {
  const float* z  = (const float*)d_in[0];
  const float* mu = (const float*)d_in[1];
  float* out = (float*)d_out;
  float* ws  = (float*)d_ws;
  const size_t NK = (size_t)NROWS * KCOLS;

  // workspace layout (floats)
  float2* zs      = (float2*)ws;                    // 2*N
  float2* ms      = (float2*)(ws + 131072);         // 2*256
  float*  relz    = ws + 131584;                    // N*5
  float*  sigz    = ws + 459264;                    // N
  float*  colpart = ws + 524800;                    // 256*256*5
  float*  relu_   = ws + 852480;                    // 256*5
  float*  sigu    = ws + 853760;                    // 256
  float*  cspart  = ws + 854016;                    // 256*256
  float*  colsum  = ws + 919552;                    // 256
  float*  dbuf    = out + 3 * NK;                   // Dmat slot doubles as d_ scratch

  stats_kernel<<<NROWS / 8, 256, 0, stream>>>(z, zs, NROWS);
  stats_kernel<<<KCOLS / 8, 256, 0, stream>>>(mu, ms, KCOLS);
  dist_wmma_kernel<<<NROWS / 16, 256, 0, stream>>>(z, mu, zs, ms, dbuf);
  rowtopk_kernel<<<NROWS / 8, 256, 0, stream>>>(dbuf, relz);
  colpart_kernel<<<256, 256, 0, stream>>>(dbuf, colpart);
  colmerge_kernel<<<1, 256, 0, stream>>>(colpart, relu_);
  sigma_kernel<<<NROWS / 256, 256, 0, stream>>>(relz, sigz, 1.3219281f, NROWS);
  sigma_kernel<<<1, 256, 0, stream>>>(relu_, sigu, 2.3219281f, KCOLS);
  ws_kernel<<<NROWS, 256, 0, stream>>>(dbuf, relz, sigz, relu_, sigu, out);
  cspart_kernel<<<256, 256, 0, stream>>>(out + NK, cspart);
  csmerge_kernel<<<1, 256, 0, stream>>>(cspart, colsum);
  dmat_kernel<<<NROWS, 256, 0, stream>>>(colsum, out);
}